// BilateralSliceApply_40183714021706
// MI455X (gfx1250) — compile-verified
//
#include <hip/hip_runtime.h>
#include <stdint.h>

// Bilateral grid slice + per-pixel affine apply (HDRNet).
// grid:  [4, 12, 8, 16, 16] f32   guide: [4, 1024, 1024] f32
// image: [4, 3, 1024, 1024] f32   out:   [4, 3, 1024, 1024] f32
//
// One workgroup = one 64x64 pixel tile of one batch (64 px == 1 grid cell).
// The 3x3-cell x 8-depth x 12-channel grid window (3456 B) is staged into LDS
// in channel-LAST layout [wy][wx][d][c] by the Tensor Data Mover (5-D
// descriptor performs the transpose during the DMA), then every pixel does
// 8 corners x 3 ds_load_b128 + fused affine accumulation.

#define Cc 12
#define Dd 8
#define GHh 16
#define GWw 16
#define Hh 1024
#define Ww 1024

#if defined(__HIP_DEVICE_COMPILE__) && __has_builtin(__builtin_amdgcn_tensor_load_to_lds)
#define USE_TDM 1
#else
#define USE_TDM 0
#endif

typedef unsigned int u32x4 __attribute__((ext_vector_type(4)));
typedef int          i32x4 __attribute__((ext_vector_type(4)));
typedef int          i32x8 __attribute__((ext_vector_type(8)));

__device__ __forceinline__ int iclamp(int v, int lo, int hi) {
    return v < lo ? lo : (v > hi ? hi : v);
}

__global__ __launch_bounds__(256) void BilateralSliceApply_kernel(
    const float* __restrict__ grid, const float* __restrict__ guide,
    const float* __restrict__ image, float* __restrict__ out)
{
    // LDS window: [wy(3)][wx(3)][d(8)][c(12)]  (channel-last, 12-dword corner rows)
    __shared__ __align__(16) float sg[3 * 3 * Dd * Cc];

    const int tX = blockIdx.x;      // tile x  (grid cell)
    const int tY = blockIdx.y;      // tile y  (grid cell)
    const int b  = blockIdx.z;      // batch
    const int wx0 = iclamp(tX - 1, 0, GWw - 3);   // window origin in grid cells
    const int wy0 = iclamp(tY - 1, 0, GHh - 3);

#if USE_TDM
    if (threadIdx.x < 32u) {        // wave 0 issues the DMA
        const uint64_t gaddr = (uint64_t)(uintptr_t)(grid
            + (size_t)b * (Cc * Dd * GHh * GWw)     // batch slab
            + (wy0 * GWw + wx0));                   // window corner (c=0,d=0)
        const uint32_t lds_base = (uint32_t)(uintptr_t)&sg[0]; // low 32b = LDS offset

        // D# group 0: count=1 | lds_addr | global_addr[56:0] | type=2
        u32x4 g0;
        g0.x = 1u;
        g0.y = lds_base;
        g0.z = (uint32_t)(gaddr & 0xFFFFFFFFull);
        g0.w = (uint32_t)((gaddr >> 32) & 0x1FFFFFFull) | (2u << 30);

        // D# group 1: data_size=4B; dims: X=1, Y=c(12, stride 2048), then
        // tile_dim2 = d(8);  tensor_dim1_stride = 256 (per-d)
        i32x8 g1 = {
            (2 << 16),                 // data_size = 4 bytes
            (1 << 16),                 // tensor_dim0 = 1
            (Cc << 16),                // tensor_dim1 = 12
            (1 << 16),                 // tile_dim0 = 1
            Cc | (Dd << 16),           // tile_dim1 = 12, tile_dim2 = 8
            Dd * GHh * GWw,            // tensor_dim0_stride = 2048 (per-c)
            ((GHh * GWw) << 16),       // tensor_dim1_stride = 256 (per-d)
            0
        };
        // D# group 2: tensor_dim2=8, tensor_dim3=3 (x cells), stride2=1 (per-x), tile_dim3=3
        i32x4 g2 = { Dd, 3, 1, (3 << 16) };
        // D# group 3: stride3=16 (per-y), tensor_dim4=3, tile_dim4=3
        i32x4 g3 = { GWw, (3 << 16), (3 << 16), 0 };

#if defined(__clang_major__) && __clang_major__ >= 23
        i32x8 gz8 = { 0, 0, 0, 0, 0, 0, 0, 0 };
        __builtin_amdgcn_tensor_load_to_lds(g0, g1, g2, g3, gz8, 0);
#else
        __builtin_amdgcn_tensor_load_to_lds(g0, g1, g2, g3, 0);
#endif
        __builtin_amdgcn_s_wait_tensorcnt(0);
    }
#else
    // Fallback staging: cooperative transpose through L2 (grid is L2-resident).
    for (int i = threadIdx.x; i < 3 * 3 * Dd * Cc; i += 256) {
        int c = i % Cc;  int t = i / Cc;
        int d = t % Dd;  t /= Dd;
        int gx = t % 3;  int gy = t / 3;
        sg[i] = grid[(((size_t)b * Cc + c) * Dd + d) * (GHh * GWw)
                     + (wy0 + gy) * GWw + (wx0 + gx)];
    }
#endif
    __syncthreads();

    // Per-thread x is fixed for all 16 row-slabs: hoist x weights/corners.
    const int tx  = threadIdx.x & 63;
    const int tyo = threadIdx.x >> 6;            // 0..3
    const int x   = tX * 64 + tx;

    const float gxc = ((float)x + 0.5f) * (1.0f / 64.0f);
    const float fxf = floorf(gxc - 0.5f);
    const float txf = gxc - 0.5f - fxf;          // in [0,1)
    const int   fxi = (int)fxf;
    const int   lxs[2] = { iclamp(fxi, 0, GWw - 1) - wx0,
                           iclamp(fxi + 1, 0, GWw - 1) - wx0 };
    const float wxs[2] = { 1.0f - txf, txf };

    const int planeHW = Hh * Ww;

    for (int r = 0; r < 16; ++r) {
        const int y = tY * 64 + r * 4 + tyo;

        const float gyc = ((float)y + 0.5f) * (1.0f / 64.0f);
        const float fyf = floorf(gyc - 0.5f);
        const float tyf = gyc - 0.5f - fyf;
        const int   fyi = (int)fyf;
        const int   lys[2] = { iclamp(fyi, 0, GHh - 1) - wy0,
                               iclamp(fyi + 1, 0, GHh - 1) - wy0 };
        const float wys[2] = { 1.0f - tyf, tyf };

        const int gidx = (b * Hh + y) * Ww + x;
        const float gzc = guide[gidx] * (float)Dd;
        const float fzf = floorf(gzc - 0.5f);
        const float tzf = gzc - 0.5f - fzf;
        const int   fzi = (int)fzf;
        const int   zs[2] = { iclamp(fzi, 0, Dd - 1), iclamp(fzi + 1, 0, Dd - 1) };
        const float wzs[2] = { 1.0f - tzf, tzf };

        const int ip = (b * 3 * Hh + y) * Ww + x;
        const float ir = image[ip];
        const float ig = image[ip + planeHW];
        const float ib = image[ip + 2 * planeHW];

        float a0 = 0.0f, a1 = 0.0f, a2 = 0.0f;
        #pragma unroll
        for (int jy = 0; jy < 2; ++jy) {
            #pragma unroll
            for (int jx = 0; jx < 2; ++jx) {
                const float wxy  = wys[jy] * wxs[jx];
                const int baseYX = (lys[jy] * 3 + lxs[jx]) * Dd;
                #pragma unroll
                for (int jz = 0; jz < 2; ++jz) {
                    const float w = wxy * wzs[jz];
                    const float4* p =
                        (const float4*)&sg[(baseYX + zs[jz]) * Cc];
                    const float4 q0 = p[0];   // channels 0..3  -> out R row
                    const float4 q1 = p[1];   // channels 4..7  -> out G row
                    const float4 q2 = p[2];   // channels 8..11 -> out B row
                    a0 = fmaf(w, fmaf(q0.x, ir, fmaf(q0.y, ig, fmaf(q0.z, ib, q0.w))), a0);
                    a1 = fmaf(w, fmaf(q1.x, ir, fmaf(q1.y, ig, fmaf(q1.z, ib, q1.w))), a1);
                    a2 = fmaf(w, fmaf(q2.x, ir, fmaf(q2.y, ig, fmaf(q2.z, ib, q2.w))), a2);
                }
            }
        }
        out[ip]                = a0;
        out[ip + planeHW]      = a1;
        out[ip + 2 * planeHW]  = a2;
    }
}

extern "C" void kernel_launch(void* const* d_in, const int* in_sizes, int n_in,
                              void* d_out, int out_size, void* d_ws, size_t ws_size,
                              hipStream_t stream) {
    (void)in_sizes; (void)n_in; (void)out_size; (void)d_ws; (void)ws_size;
    const float* grid  = (const float*)d_in[0];
    const float* guide = (const float*)d_in[1];
    const float* image = (const float*)d_in[2];
    float* out = (float*)d_out;

    dim3 g(GWw, GHh, 4);   // 16 x 16 tiles x 4 batches
    dim3 blk(256);
    hipLaunchKernelGGL(BilateralSliceApply_kernel, g, blk, 0, stream,
                       grid, guide, image, out);
}